// generate_graph_86071144612517
// MI455X (gfx1250) — compile-verified
//
#include <hip/hip_runtime.h>
#include <math.h>

// Problem constants (from reference)
#define NPTS 8192
#define CDIM 256
#define DDIM 10
#define DPAD 16      // emb padded to 16 so K fits exactly 4 WMMA f32 16x16x4 chunks
#define KSEL 16
#define EPS_BN 1e-5f
#define INV_TEMP 2.0f   // 1 / TEMP, TEMP = 0.5

typedef __attribute__((ext_vector_type(2))) float v2f;
typedef __attribute__((ext_vector_type(8))) float v8f;

// ---------------------------------------------------------------------------
// Per-lane sorted top-K insertion (fully unrolled -> stays in VGPRs).
// DESC=true keeps largest-first; DESC=false keeps smallest-first.
// ---------------------------------------------------------------------------
template <bool DESC>
__device__ inline void topk_insert(float (&tv)[KSEL], int (&tix)[KSEL], float v, int idx) {
  bool better = DESC ? (v > tv[KSEL - 1]) : (v < tv[KSEL - 1]);
  if (!better) return;
  float cv = v; int ci = idx;
#pragma unroll
  for (int k = 0; k < KSEL; ++k) {
    bool sw = DESC ? (cv > tv[k]) : (cv < tv[k]);
    float ov = tv[k]; int oi = tix[k];
    tv[k] = sw ? cv : ov; tix[k] = sw ? ci : oi;
    cv = sw ? ov : cv;   ci = sw ? oi : ci;
  }
}

// ---------------------------------------------------------------------------
// Wave-wide (32-lane) merge of per-lane sorted top-K lists via a 16-round
// argmax tournament on a packed u64 key = {sortable_bits, idx<<5 | lane}.
// Values are nonnegative floats so the raw bit pattern is order-preserving
// (inverted for ascending). After the call lane t (t<16) holds the t-th best.
// ---------------------------------------------------------------------------
template <bool DESC>
__device__ inline void wave_topk_merge(float (&tv)[KSEL], int (&tix)[KSEL], int lane,
                                       float &out_v, int &out_i) {
  int popped = 0;
  out_v = 0.f; out_i = 0;
#pragma unroll 1
  for (int t = 0; t < KSEL; ++t) {
    unsigned kb = DESC ? __float_as_uint(tv[0]) : ~__float_as_uint(tv[0]);
    if (popped >= KSEL) kb = 0u;
    unsigned long long key = ((unsigned long long)kb << 32) |
                             (unsigned)(((unsigned)tix[0] << 5) | (unsigned)lane);
#pragma unroll
    for (int m = 16; m >= 1; m >>= 1) {
      unsigned long long o = __shfl_xor(key, m, 32);
      key = (o > key) ? o : key;
    }
    int wlane = (int)(key & 31ull);
    int widx  = (int)((key >> 5) & 0x1FFFull);
    unsigned wb = (unsigned)(key >> 32);
    float wv = DESC ? __uint_as_float(wb) : __uint_as_float(~wb);
    if (lane == wlane) {           // winner pops its head (constant-index shift)
      popped++;
#pragma unroll
      for (int k = 0; k < KSEL - 1; ++k) { tv[k] = tv[k + 1]; tix[k] = tix[k + 1]; }
      tv[KSEL - 1] = DESC ? -3.0e38f : 3.0e38f;
      tix[KSEL - 1] = 0;
    }
    if (lane == t) { out_v = wv; out_i = widx; }
  }
}

// ---------------------------------------------------------------------------
// Kernel 1: h = x @ W1 + b1 via V_WMMA_F32_16X16X4_F32.
// One wave per 16x16 tile; K loop = 64 chunks of 4.
// A layout: lanes 0-15 hold (M=lane, K=k..k+1), lanes 16-31 hold (M, K=k+2..k+3).
// B layout mirrors A with N in place of M. C/D: VGPR r -> M=r (+8 for hi lanes).
// ---------------------------------------------------------------------------
__global__ void gemm1_wmma(const float* __restrict__ x, const float* __restrict__ W1,
                           const float* __restrict__ b1, float* __restrict__ h) {
  const int lane = threadIdx.x & 31;
  const int wave = (blockIdx.x * blockDim.x + threadIdx.x) >> 5;
  const int m0 = (wave >> 4) * 16;
  const int n0 = (wave & 15) * 16;
  const int half = lane >> 4;
  const int l16  = lane & 15;
  const int arow = m0 + l16;
  const int bcol = n0 + l16;

  v8f c = {};
#pragma unroll 4
  for (int k = 0; k < CDIM; k += 4) {
    int ka = k + half * 2;
    v2f a, b;
    a.x = x[arow * CDIM + ka];
    a.y = x[arow * CDIM + ka + 1];
    b.x = W1[ka * CDIM + bcol];
    b.y = W1[(ka + 1) * CDIM + bcol];
    c = __builtin_amdgcn_wmma_f32_16x16x4_f32(false, a, false, b, (short)0, c, false, false);
  }
  float bias = b1[bcol];
#pragma unroll
  for (int r = 0; r < 8; ++r) {
    int m = m0 + r + half * 8;
    h[m * CDIM + bcol] = c[r] + bias;
  }
}

// ---------------------------------------------------------------------------
// Kernel 2: per-channel BN train stats -> scale/shift. One block per channel.
// ---------------------------------------------------------------------------
__global__ void bn_stats(const float* __restrict__ h, const float* __restrict__ gamma,
                         const float* __restrict__ beta, float* __restrict__ scale,
                         float* __restrict__ shiftv) {
  __shared__ float s_sum[256], s_sq[256];
  int c = blockIdx.x, t = threadIdx.x;
  float s = 0.f, q = 0.f;
  for (int r = t; r < NPTS; r += 256) {
    float v = h[r * CDIM + c];
    s += v; q += v * v;
  }
  s_sum[t] = s; s_sq[t] = q;
  __syncthreads();
  for (int off = 128; off > 0; off >>= 1) {
    if (t < off) { s_sum[t] += s_sum[t + off]; s_sq[t] += s_sq[t + off]; }
    __syncthreads();
  }
  if (t == 0) {
    float mu  = s_sum[0] * (1.0f / NPTS);
    float var = s_sq[0] * (1.0f / NPTS) - mu * mu;
    float sc  = gamma[c] * rsqrtf(var + EPS_BN);
    scale[c]  = sc;
    shiftv[c] = beta[c] - mu * sc;
  }
}

// ---------------------------------------------------------------------------
// Kernel 3: emb = relu(BN(h)) @ W2 + b2 + 0.001*noise, padded to 16 cols,
// plus per-row squared norms. 16 rows x 16 d-lanes per block.
// ---------------------------------------------------------------------------
__global__ void mlp2_emb(const float* __restrict__ h, const float* __restrict__ scale,
                         const float* __restrict__ shiftv, const float* __restrict__ W2,
                         const float* __restrict__ b2, const float* __restrict__ noise,
                         float* __restrict__ embP, float* __restrict__ sqn) {
  int t = threadIdx.x;
  int d = t & 15;
  int n = blockIdx.x * 16 + (t >> 4);
  float acc = (d < DDIM) ? (b2[d] + 0.001f * noise[n * DDIM + d]) : 0.f;
  for (int c = 0; c < CDIM; ++c) {
    float a = fmaxf(fmaf(h[n * CDIM + c], scale[c], shiftv[c]), 0.f);
    if (d < DDIM) acc = fmaf(a, W2[c * DDIM + d], acc);
  }
  embP[n * DPAD + d] = acc;
  float s = acc * acc;           // reduce within 16-lane group (wave32)
  s += __shfl_xor(s, 1, 32);
  s += __shfl_xor(s, 2, 32);
  s += __shfl_xor(s, 4, 32);
  s += __shfl_xor(s, 8, 32);
  if (d == 0) sqn[n] = s;
}

// ---------------------------------------------------------------------------
// Kernel 4: KNN(k=16) on 3-D positions, one wave per row, lane-strided scan
// with register top-K + wave tournament merge. pos (96 KB) lives in cache.
// ---------------------------------------------------------------------------
__global__ void pos_knn(const float* __restrict__ pos,
                        int* __restrict__ e0k, int* __restrict__ e1k) {
  const int lane = threadIdx.x & 31;
  const int i = (blockIdx.x * blockDim.x + threadIdx.x) >> 5;
  float px = pos[i * 3], py = pos[i * 3 + 1], pz = pos[i * 3 + 2];
  float tv[KSEL]; int tix[KSEL];
#pragma unroll
  for (int k = 0; k < KSEL; ++k) { tv[k] = 3.0e38f; tix[k] = 0; }
  for (int j = lane; j < NPTS; j += 32) {
    float dx = px - pos[j * 3];
    float dy = py - pos[j * 3 + 1];
    float dz = pz - pos[j * 3 + 2];
    float d2 = fmaf(dx, dx, fmaf(dy, dy, dz * dz));
    if (j == i) d2 = 3.0e38f;                    // loop=False: exclude self
    topk_insert<false>(tv, tix, d2, j);
  }
  float ov; int oi;
  wave_topk_merge<false>(tv, tix, lane, ov, oi);
  if (lane < KSEL) {
    e0k[i * KSEL + lane] = oi;
    e1k[i * KSEL + lane] = i;
  }
}

// ---------------------------------------------------------------------------
// Kernel 5: column-wise (axis 0) online softmax stats over noisy_logits/TEMP,
// never materializing the 256 MB matrix. One wave owns a 16-column strip,
// holds the B operand (emb rows j0..j0+15, K=16) in registers, and walks all
// 512 i-tiles; each tile is 4x V_WMMA_F32_16X16X4_F32 producing G = emb_i.emb_j.
// Column reduction = 8-reg reduce + shfl_xor(16) (C layout: lane n / lane n+16
// hold rows m and m+8 of column n).
// ---------------------------------------------------------------------------
__global__ void col_softmax_stats(const float* __restrict__ embP, const float* __restrict__ sqn,
                                  const float* __restrict__ gu,
                                  float* __restrict__ colmax, float* __restrict__ colsum) {
  const int lane = threadIdx.x & 31;
  const int wave = (blockIdx.x * blockDim.x + threadIdx.x) >> 5;
  const int j0 = wave * 16;
  const int half = lane >> 4;
  const int l16  = lane & 15;
  const int jcol = j0 + l16;

  v2f b[4];                       // B[k,n] = embP[j0+n][k], resident for whole loop
#pragma unroll
  for (int c = 0; c < 4; ++c) {
    int kk = c * 4 + half * 2;
    b[c].x = embP[jcol * DPAD + kk];
    b[c].y = embP[jcol * DPAD + kk + 1];
  }
  float sqj  = sqn[jcol];
  float runm = -3.0e38f, runs = 0.f;

  for (int i0 = 0; i0 < NPTS; i0 += 16) {
    v8f cacc = {};
    const int irow = i0 + l16;
#pragma unroll
    for (int c = 0; c < 4; ++c) {
      int kk = c * 4 + half * 2;
      v2f a;
      a.x = embP[irow * DPAD + kk];
      a.y = embP[irow * DPAD + kk + 1];
      cacc = __builtin_amdgcn_wmma_f32_16x16x4_f32(false, a, false, b[c], (short)0, cacc,
                                                   false, false);
    }
    float lt[8];
    float tmax = -3.0e38f;
#pragma unroll
    for (int r = 0; r < 8; ++r) {
      int ig = i0 + r + half * 8;
      float d2 = fmaxf(sqn[ig] + sqj - 2.f * cacc[r], 0.f);
      float p = __expf(-d2);
      float u = gu[(long long)ig * NPTS + jcol];
      float g = -__logf(-__logf(u + 1e-20f) + 1e-20f);
      float l = (__logf(p + 1e-20f) + g) * INV_TEMP;
      lt[r] = l;
      tmax = fmaxf(tmax, l);
    }
    tmax = fmaxf(tmax, __shfl_xor(tmax, 16, 32));   // full 16-row tile max
    float newm = fmaxf(runm, tmax);
    float tsum = 0.f;
#pragma unroll
    for (int r = 0; r < 8; ++r) tsum += __expf(lt[r] - newm);
    tsum += __shfl_xor(tsum, 16, 32);
    runs = runs * __expf(runm - newm) + tsum;       // online rescale
    runm = newm;
  }
  if (lane < 16) {
    colmax[jcol] = runm;
    colsum[jcol] = runs;
  }
}

// ---------------------------------------------------------------------------
// Kernel 6: row-wise top-16 of probs[i,j] = exp(l/T - colmax[j]) / colsum[j].
// One wave per row; emb row i cached in 16 VGPRs; lane-strided j scan,
// register top-K + tournament merge. Writes top_v/top_i and edge rows.
// ---------------------------------------------------------------------------
__global__ void row_topk(const float* __restrict__ embP, const float* __restrict__ sqn,
                         const float* __restrict__ gu, const float* __restrict__ colmax,
                         const float* __restrict__ colsum,
                         float* __restrict__ top_v, int* __restrict__ top_i,
                         int* __restrict__ e0, int* __restrict__ e1) {
  const int lane = threadIdx.x & 31;
  const int i = (blockIdx.x * blockDim.x + threadIdx.x) >> 5;
  float e[DPAD];
#pragma unroll
  for (int d = 0; d < DPAD; ++d) e[d] = embP[i * DPAD + d];
  float sqi = sqn[i];
  float tv[KSEL]; int tix[KSEL];
#pragma unroll
  for (int k = 0; k < KSEL; ++k) { tv[k] = -3.0e38f; tix[k] = 0; }
  const long long rowoff = (long long)i * NPTS;
  for (int j = lane; j < NPTS; j += 32) {
    float dot = 0.f;
#pragma unroll
    for (int d = 0; d < DPAD; ++d) dot = fmaf(e[d], embP[j * DPAD + d], dot);
    float d2 = fmaxf(sqi + sqn[j] - 2.f * dot, 0.f);
    float p = __expf(-d2);
    float u = gu[rowoff + j];
    float g = -__logf(-__logf(u + 1e-20f) + 1e-20f);
    float l = (__logf(p + 1e-20f) + g) * INV_TEMP;
    float prob = __expf(l - colmax[j]) / colsum[j];
    topk_insert<true>(tv, tix, prob, j);
  }
  float ov; int oi;
  wave_topk_merge<true>(tv, tix, lane, ov, oi);
  if (lane < KSEL) {
    top_v[i * KSEL + lane] = ov;
    top_i[i * KSEL + lane] = oi;
    e0[i * KSEL + lane] = oi;     // soft edges occupy first N*K slots of row 0
    e1[i * KSEL + lane] = i;
  }
}

// ---------------------------------------------------------------------------
// Launch: GEMM1 -> BN stats -> emb (+sqnorm) -> [pos KNN] -> col softmax
// stats -> row top-k. All on `stream`; scratch carved from d_ws (~9 MB).
// ---------------------------------------------------------------------------
extern "C" void kernel_launch(void* const* d_in, const int* in_sizes, int n_in,
                              void* d_out, int out_size, void* d_ws, size_t ws_size,
                              hipStream_t stream) {
  (void)in_sizes; (void)n_in; (void)out_size; (void)ws_size;
  const float* x     = (const float*)d_in[0];
  const float* pos   = (const float*)d_in[1];
  const float* W1    = (const float*)d_in[2];
  const float* b1    = (const float*)d_in[3];
  const float* gamma = (const float*)d_in[4];
  const float* beta  = (const float*)d_in[5];
  const float* W2    = (const float*)d_in[6];
  const float* b2    = (const float*)d_in[7];
  const float* noise = (const float*)d_in[8];
  const float* gu    = (const float*)d_in[9];

  float* ws     = (float*)d_ws;
  float* h      = ws;                      // 8192*256
  float* scale  = h + NPTS * CDIM;         // 256
  float* shiftv = scale + CDIM;            // 256
  float* embP   = shiftv + CDIM;           // 8192*16
  float* sqn    = embP + NPTS * DPAD;      // 8192
  float* colmax = sqn + NPTS;              // 8192
  float* colsum = colmax + NPTS;           // 8192

  // d_out: top_v [N*K] f32 | top_i [N*K] i32 | edge_index [2, 2*N*K] i32
  float* top_v = (float*)d_out;
  int*   top_i = (int*)((float*)d_out + NPTS * KSEL);
  int*   e0    = (int*)((float*)d_out + 2 * NPTS * KSEL);   // edge row 0
  int*   e1    = e0 + 2 * NPTS * KSEL;                      // edge row 1

  gemm1_wmma       <<<1024, 256, 0, stream>>>(x, W1, b1, h);
  bn_stats         <<<CDIM, 256, 0, stream>>>(h, gamma, beta, scale, shiftv);
  mlp2_emb         <<<NPTS / 16, 256, 0, stream>>>(h, scale, shiftv, W2, b2, noise, embP, sqn);
  pos_knn          <<<NPTS / 8, 256, 0, stream>>>(pos, e0 + NPTS * KSEL, e1 + NPTS * KSEL);
  col_softmax_stats<<<(NPTS / 16) / 8, 256, 0, stream>>>(embP, sqn, gu, colmax, colsum);
  row_topk         <<<NPTS / 8, 256, 0, stream>>>(embP, sqn, gu, colmax, colsum,
                                                  top_v, top_i, e0, e1);
}